// MultiHashPyramid_42606075576717
// MI455X (gfx1250) — compile-verified
//
#include <hip/hip_runtime.h>
#include <hip/hip_bf16.h>

// ---------------------------------------------------------------------------
// MultiHashPyramid for MI455X (gfx1250, wave32).
//
// Memory-bound gather kernel: 67MB table is L2-resident (192MB L2), output is
// 134MB streamed -> ~9us HBM floor. The only GEMM-shaped work (512x8 logit
// matvec -> sign bits) runs on V_WMMA_F32_16X16X4_F32 (exact fp32 matrix op),
// one wave per 16-token tile, 4x ds_load_2addr_b64 + 4 wmma per unrolled step.
// ---------------------------------------------------------------------------

typedef __attribute__((ext_vector_type(2))) float v2f;
typedef __attribute__((ext_vector_type(8))) float v8f;

#define NSCALE   4
#define KHASH    4
#define BUCKETS  16384
#define EDIM     64
#define TLEN     4096
#define TILE     16          // tokens per workgroup (one WMMA M-tile)
#define SROW     516         // 512 features + 4 floats pad (LDS bank spread)

__device__ __constant__ unsigned c_primes[16] = {
    1000003u, 999983u, 1299709u, 15485863u, 32452843u, 49979687u,
    67867967u, 86028121u, 104395301u, 122949823u, 141650939u, 160481183u,
    179424673u, 198491317u, 217645177u, 236887691u};
__device__ __constant__ unsigned c_salts[4] = {
    0u, 3735928559u, 3405691582u, 2343432205u};
__device__ __constant__ unsigned c_condp[8] = {
    2654435761u, 2246822519u, 3266489917u, 2028178513u,
    1220703125u, 1610612741u, 805306457u, 402653189u};

__launch_bounds__(128)
__global__ void mh_pyramid_kernel(const int*   __restrict__ tokens,
                                  const float* __restrict__ tables,
                                  const float* __restrict__ condW,
                                  float*       __restrict__ out)
{
    __shared__ __align__(16) float  s_short[TILE * SROW]; // 33.0 KB short features
    // s_cw2[p][n] = (cond_W[n][2p], cond_W[n][2p+1]), p = feature-pair 0..255,
    // n = output column 0..7.  16 KB.
    __shared__ __align__(16) float2 s_cw2[256 * 8];
    __shared__ unsigned             s_keys[16 * TILE];    // keys[(s*4+k)][m]
    __shared__ unsigned             s_tok[32];            // tokens t0-16 .. t0+15
    __shared__ unsigned             s_ckey[TILE];         // cond_key per token

    const int tid  = threadIdx.x;
    const int blk  = blockIdx.x;
    const int brow = blk >> 8;            // batch row (4096/16 = 256 tiles/row)
    const int t0   = (blk & 255) * TILE;  // first token of tile

    // ---- Stage 0: token window, cond_W pair-transpose into LDS, zero ckeys
    if (tid < 32) {
        int t = t0 - 16 + tid;
        s_tok[tid] = (t >= 0) ? (unsigned)tokens[brow * TLEN + t] : 0u;
    }
    if (tid < TILE) s_ckey[tid] = 0u;
    for (int j = tid; j < 256 * 8; j += 128) {
        const int p = j >> 3, n = j & 7;
        s_cw2[j] = make_float2(condW[n * 512 + 2 * p], condW[n * 512 + 2 * p + 1]);
    }
    __syncthreads();

    // ---- Stage 1: hash keys, 2 (s,k) pairs per thread, int32 wraparound math
    {
        const int m = tid & 15;           // token within tile
        const int j = tid >> 4;           // 0..7
        #pragma unroll
        for (int half = 0; half < 2; ++half) {
            const int idx = j + half * 8; // idx = s*4 + k, 0..15
            const int s = idx >> 2, k = idx & 3;
            const int w = 2 << s;         // window 2,4,8,16
            const unsigned salt = c_salts[k];
            unsigned h = 0u;
            for (int i = 0; i < w; ++i)   // padded tokens are 0 -> contribute 0
                h ^= s_tok[16 + m - 1 - i] * (c_primes[i] ^ salt);
            s_keys[idx * TILE + m] = h & 16383u;   // floored mod 16384
        }
    }
    __syncthreads();

    const size_t outTokBase = (size_t)(brow * TLEN + t0) * 1024;

    // ---- Stage 2: short-scale gathers (s=0,1): 128 rows x 16 float4 each.
    // 16 consecutive lanes fetch one coalesced 256B table row.
    for (int r = 0; r < 16; ++r) {
        const int flat = r * 128 + tid;
        const int row  = flat >> 4;       // 0..127
        const int q    = flat & 15;       // float4 slot within row
        const int s    = row >> 6;        // 0..1
        const int k    = (row >> 4) & 3;
        const int m    = row & 15;
        const int idx  = s * 4 + k;
        const unsigned key = s_keys[idx * TILE + m];
        const float4 v = *(const float4*)(tables + ((size_t)idx * BUCKETS + key) * EDIM + q * 4);
        const int d = s * 256 + k * 64 + q * 4;
        *(float4*)(out + outTokBase + (size_t)m * 1024 + d) = v;
        *(float4*)(&s_short[m * SROW + d]) = v;
    }
    __syncthreads();

    // ---- Stage 3: logits via V_WMMA_F32_16X16X4_F32 on wave 0.
    // A = 16 tokens x 512 feats (K-chunks of 4), B = cond_W^T (4 x 16).
    // A layout: lanes 0-15 -> M=lane, vgpr{0,1}=K{0,1}; lanes 16-31 -> K{2,3}.
    // B layout: lanes 0-15 -> N=lane, vgpr{0,1}=K{0,1}; lanes 16-31 -> K{2,3}.
    // B columns N>=8 only feed C columns N>=8 which we never consume, so those
    // lanes load duplicate data (ln&7) unconditionally -- no EXEC masking.
    if (tid < 32) {
        const int lane = tid;
        const int hi   = lane >> 4;       // 0 -> K pair {0,1}, 1 -> {2,3}
        const int ln   = lane & 15;       // M (for A) / N (for B)
        const float*  aP = &s_short[ln * SROW + hi * 2];
        const float2* bP = &s_cw2[hi * 8 + (ln & 7)];
        v8f c = {};
        #pragma unroll 4
        for (int kk = 0; kk < 128; ++kk) {
            const v2f a = *(const v2f*)(aP + kk * 4);        // ds_load_b64
            const float2 b2 = bP[kk * 16];                   // ds_load_b64
            v2f b; b.x = b2.x; b.y = b2.y;
            c = __builtin_amdgcn_wmma_f32_16x16x4_f32(
                    false, a, false, b, (short)0, c, false, false);
        }
        // C layout: vgpr r -> M = r + 8*hi, N = ln. Fold sign bits into
        // cond_key branchlessly: xor-with-0 is a no-op, so ds_xor issues
        // unconditionally; prime hoisted to a register (single load).
        if (ln < 8) {
            const unsigned myp = c_condp[ln];
            #pragma unroll
            for (int r = 0; r < 8; ++r)
                atomicXor(&s_ckey[r + hi * 8], (c[r] > 0.0f) ? myp : 0u);
        }
    }
    __syncthreads();

    // ---- Stage 4: long-scale gathers (s=2,3) with re-keyed indices.
    for (int r = 0; r < 16; ++r) {
        const int flat = r * 128 + tid;
        const int row  = flat >> 4;
        const int q    = flat & 15;
        const int s    = 2 + (row >> 6);
        const int k    = (row >> 4) & 3;
        const int m    = row & 15;
        const int idx  = s * 4 + k;
        const unsigned key = (s_keys[idx * TILE + m] ^ s_ckey[m]) & 16383u;
        const float4 v = *(const float4*)(tables + ((size_t)idx * BUCKETS + key) * EDIM + q * 4);
        const int d = s * 256 + k * 64 + q * 4;
        *(float4*)(out + outTokBase + (size_t)m * 1024 + d) = v;
    }
}

extern "C" void kernel_launch(void* const* d_in, const int* in_sizes, int n_in,
                              void* d_out, int out_size, void* d_ws, size_t ws_size,
                              hipStream_t stream) {
    (void)in_sizes; (void)n_in; (void)d_ws; (void)ws_size; (void)out_size;
    const int*   tokens = (const int*)  d_in[0];   // (8, 4096) int32
    const float* tables = (const float*)d_in[1];   // (16, 16384, 64) f32
    const float* condW  = (const float*)d_in[2];   // (8, 512) f32
    float*       out    = (float*)d_out;           // (8, 4096, 1024) f32

    const int nTiles = (8 * TLEN) / TILE;          // 2048 workgroups
    mh_pyramid_kernel<<<nTiles, 128, 0, stream>>>(tokens, tables, condW, out);
}